// AttentionSE3_22290880266598
// MI455X (gfx1250) — compile-verified
//
#include <hip/hip_runtime.h>

// ---------------------------------------------------------------------------
// SE(3) attention layer for MI455X (gfx1250, wave32, WMMA).
//
// Structure:
//   conv_kernel : one block per (node, conv∈{k,v}); 256 threads = 8 waves.
//     - gathers neighbor fibers, builds T[p,i,f] tensors in LDS
//     - per net: trunk MLP (layer2 via V_WMMA_F32_16X16X4_F32),
//       then fused  R = h2@w3  GEMM (WMMA) + R·T contraction -> k/v in d_ws
//   attn_kernel : one block per node; masked softmax attention + out proj.
// ---------------------------------------------------------------------------

namespace {

constexpr int BATCH = 2;
constexpr int NN    = 128;
constexpr int JJ    = 16;
constexpr int DIMC  = 16;
constexpr int HIDC  = 128;
constexpr int MIDC  = 128;
constexpr int NODES = BATCH * NN;   // 256

// workspace layout (float offsets)
constexpr long K0_OFF = 0;                                   // [node][j][HID]
constexpr long K1_OFF = (long)NODES * JJ * HIDC;             // [node][j][HID][3]
constexpr long V0_OFF = K1_OFF + (long)NODES * JJ * HIDC * 3;
constexpr long V1_OFF = V0_OFF + (long)NODES * JJ * HIDC;

typedef __attribute__((ext_vector_type(2))) float v2f;
typedef __attribute__((ext_vector_type(8))) float v8f;

struct ConvW {
  const float *w1, *b1, *g1, *be1;
  const float *w2, *b2, *g2, *be2;
  const float *w3a, *b3a, *w3b, *b3b;
};

// C(16x16) += A(16xK, rows in LDS) * B(KxN16, global, leading dim ldb).
// f32 WMMA 16x16x4: A lane layout  lane<16: M=lane, K {k0,k0+1};
//                                  lane>=16: M=lane-16, K {k0+2,k0+3}.
// B mirrors A with N in place of M.
__device__ __forceinline__ v8f wmma16x16_f32(const float* Alds, int lda,
                                             const float* __restrict__ Bg,
                                             long ldb, int K, int lane) {
  const int mn   = lane & 15;
  const int koff = (lane >> 4) << 1;   // 0 or 2
  v8f c = {};
  for (int k0 = 0; k0 < K; k0 += 4) {
    v2f a, b;
    a.x = Alds[mn * lda + k0 + koff];
    a.y = Alds[mn * lda + k0 + koff + 1];
    b.x = Bg[(long)(k0 + koff) * ldb + mn];
    b.y = Bg[(long)(k0 + koff + 1) * ldb + mn];
    c = __builtin_amdgcn_wmma_f32_16x16x4_f32(false, a, false, b,
                                              (short)0, c, false, false);
  }
  return c;
}

// D layout: VGPR r -> row (r + 8*(lane>=16)), col = lane&15.
__device__ __forceinline__ void store_tile(float* C, int ldc, v8f c, int lane) {
  const int n  = lane & 15;
  const int mb = (lane >> 4) << 3;
#pragma unroll
  for (int r = 0; r < 8; ++r) C[(mb + r) * ldc + n] = c[r];
}

// Radial trunk: h1 = relu(LN(feat*w1+b1)); h2 = relu(LN(h1@w2+b2)).
__device__ __forceinline__ void trunk(int tid, int lane, int wv,
                                      const float* sFeat,
                                      float (*sH1)[MIDC], float (*sH2)[MIDC],
                                      float* sMu, float* sRs,
                                      const ConvW& W, int net) {
  __syncthreads();
  for (int idx = tid; idx < JJ * MIDC; idx += 256) {
    const int e = idx >> 7, m = idx & 127;
    sH1[e][m] = sFeat[e] * W.w1[net * MIDC + m] + W.b1[net * MIDC + m];
  }
  __syncthreads();
  if (tid < JJ) {
    float mu = 0.f;
    for (int m = 0; m < MIDC; ++m) mu += sH1[tid][m];
    mu *= (1.f / MIDC);
    float var = 0.f;
    for (int m = 0; m < MIDC; ++m) { float d = sH1[tid][m] - mu; var += d * d; }
    sMu[tid] = mu;
    sRs[tid] = rsqrtf(var * (1.f / MIDC) + 1e-5f);
  }
  __syncthreads();
  for (int idx = tid; idx < JJ * MIDC; idx += 256) {
    const int e = idx >> 7, m = idx & 127;
    float h = (sH1[e][m] - sMu[e]) * sRs[e] * W.g1[net * MIDC + m] +
              W.be1[net * MIDC + m];
    sH1[e][m] = h > 0.f ? h : 0.f;
  }
  __syncthreads();
  // layer 2: (16 x 128) @ (128 x 128); wave wv produces column tile wv.
  {
    v8f c = wmma16x16_f32(&sH1[0][0], MIDC,
                          W.w2 + (long)net * MIDC * MIDC + 16 * wv, MIDC,
                          MIDC, lane);
    store_tile(&sH2[0][16 * wv], MIDC, c, lane);
  }
  __syncthreads();
  for (int idx = tid; idx < JJ * MIDC; idx += 256) {
    const int e = idx >> 7, m = idx & 127;
    sH2[e][m] += W.b2[net * MIDC + m];
  }
  __syncthreads();
  if (tid < JJ) {
    float mu = 0.f;
    for (int m = 0; m < MIDC; ++m) mu += sH2[tid][m];
    mu *= (1.f / MIDC);
    float var = 0.f;
    for (int m = 0; m < MIDC; ++m) { float d = sH2[tid][m] - mu; var += d * d; }
    sMu[tid] = mu;
    sRs[tid] = rsqrtf(var * (1.f / MIDC) + 1e-5f);
  }
  __syncthreads();
  for (int idx = tid; idx < JJ * MIDC; idx += 256) {
    const int e = idx >> 7, m = idx & 127;
    float h = (sH2[e][m] - sMu[e]) * sRs[e] * W.g2[net * MIDC + m] +
              W.be2[net * MIDC + m];
    sH2[e][m] = h > 0.f ? h : 0.f;
  }
  __syncthreads();
}

// Fused  R = h2 @ w3  (WMMA) then  out[e,o,p] += sum_{i,f} (R+b3)*T[e,p,i,f].
// NET 0: T00 (P=1)  NET 1: T01 (P=3)  NET 2: T10 (P=1)  NET 3: T11 (P=3,F=3)
template <int NET>
__device__ __forceinline__ void gemm_contract(
    int lane, int wv, const float (*sH2)[MIDC], float (*sCw)[16],
    const float* __restrict__ W3, const float* __restrict__ B3,
    const float (*sT00)[16], const float (*sT01)[3][16],
    const float (*sT10)[16], const float (*sT11)[3][16][3],
    float (*sOut)[384]) {
  constexpr int F      = (NET == 3) ? 3 : 1;
  constexpr int P      = (NET == 1 || NET == 3) ? 3 : 1;
  constexpr int NCOLS  = HIDC * DIMC * F;   // 2048 or 6144
  constexpr int NTILES = NCOLS / 16;        // 128 or 384
  constexpr int TPW    = NTILES / 8;        // 16 or 48 (divisible by 3)
  for (int tl = 0; tl < TPW; ++tl) {
    const int t = wv * TPW + tl;            // contiguous -> no cross-wave races
    v8f c = wmma16x16_f32(&sH2[0][0], MIDC, W3 + t * 16, NCOLS, MIDC, lane);
    store_tile(&sCw[0][0], 16, c, lane);
    __builtin_amdgcn_wave_barrier();        // order LDS store -> LDS read in-wave
    if (lane < 16) {
      const int e = lane;
#pragma unroll
      for (int p = 0; p < P; ++p) {
        float acc = 0.f;
#pragma unroll
        for (int cc = 0; cc < 16; ++cc) {
          const float val = sCw[e][cc] + B3[t * 16 + cc];
          float tv;
          if (NET == 0)      tv = sT00[e][cc];
          else if (NET == 1) tv = sT01[e][p][cc];
          else if (NET == 2) tv = sT10[e][cc];
          else {
            const int rem = (t % 3) * 16 + cc;   // col = o*48 + i*3 + f
            tv = sT11[e][p][rem / 3][rem % 3];
          }
          acc += val * tv;
        }
        const int o = (NET == 3) ? (t / 3) : t;
        sOut[e][o * P + p] += acc;
      }
    }
    __builtin_amdgcn_wave_barrier();
  }
}

__global__ __launch_bounds__(256) void conv_kernel(
    const float* __restrict__ x0, const float* __restrict__ x1,
    const int* __restrict__ nidx, const float* __restrict__ rel,
    const float* __restrict__ b00, const float* __restrict__ b01,
    const float* __restrict__ b10, const float* __restrict__ b11,
    ConvW Wk, ConvW Wv, float* __restrict__ ws) {
  const int conv = blockIdx.x & 1;
  const int node = blockIdx.x >> 1;       // b*N + n
  const int tid  = threadIdx.x;
  const int lane = tid & 31;
  const int wv   = tid >> 5;
  const int bb   = node / NN;

  __shared__ float sT00[JJ][16];
  __shared__ float sT01[JJ][3][16];
  __shared__ float sT10[JJ][16];
  __shared__ float sT11[JJ][3][16][3];
  __shared__ float sH1[JJ][MIDC];
  __shared__ float sH2[JJ][MIDC];
  __shared__ float sC[8][16][16];
  __shared__ float sOut[JJ][384];         // also gather staging at start
  __shared__ float sMu[JJ], sRs[JJ];
  __shared__ float sFeat[JJ];

  float* sX0 = &sOut[0][0];   // [16][16]     gathered x0
  float* sX1 = sX0 + 256;     // [16][16][3]  gathered x1

  // ---- gather neighbor fibers -------------------------------------------
  for (int idx = tid; idx < 256; idx += 256) {
    const int e = idx >> 4, i = idx & 15;
    const int nb = nidx[node * JJ + e];
    sX0[idx] = x0[(bb * NN + nb) * DIMC + i];
  }
  for (int idx = tid; idx < 768; idx += 256) {
    const int e = idx / 48, r = idx % 48;     // r = i*3 + m (contiguous)
    const int nb = nidx[node * JJ + e];
    sX1[idx] = x1[((bb * NN + nb) * DIMC) * 3 + r];
  }
  if (tid < JJ) sFeat[tid] = rel[node * JJ + tid];
  __syncthreads();

  // ---- T[p,i,f] = sum_q basis[p,q,f] * xg[i,q] --------------------------
  for (int idx = tid; idx < 256; idx += 256) {
    const int e = idx >> 4, i = idx & 15;
    sT00[e][i] = b00[node * JJ + e] * sX0[e * 16 + i];
  }
  for (int idx = tid; idx < 768; idx += 256) {
    const int e = idx / 48, r = idx % 48, p = r / 16, i = r % 16;
    sT01[e][p][i] = b01[(node * JJ + e) * 3 + p] * sX0[e * 16 + i];
  }
  for (int idx = tid; idx < 256; idx += 256) {
    const int e = idx >> 4, i = idx & 15;
    float s = 0.f;
    for (int q = 0; q < 3; ++q)
      s += b10[(node * JJ + e) * 3 + q] * sX1[(e * 16 + i) * 3 + q];
    sT10[e][i] = s;
  }
  for (int idx = tid; idx < 2304; idx += 256) {
    const int e = idx / 144, r = idx % 144;
    const int p = r / 48, r2 = r % 48, i = r2 / 3, f = r2 % 3;
    float s = 0.f;
    for (int q = 0; q < 3; ++q)
      s += b11[(((node * JJ + e) * 3 + p) * 3 + q) * 3 + f] *
           sX1[(e * 16 + i) * 3 + q];
    sT11[e][p][i][f] = s;
  }
  __syncthreads();

  const ConvW W = conv ? Wv : Wk;
  float* o0g = ws + (conv ? V0_OFF : K0_OFF);
  float* o1g = ws + (conv ? V1_OFF : K1_OFF);

  // ---- degree-0 output: nets (0,0) and (1,0) ----------------------------
  for (int idx = tid; idx < JJ * HIDC; idx += 256) sOut[idx >> 7][idx & 127] = 0.f;
  trunk(tid, lane, wv, sFeat, sH1, sH2, sMu, sRs, W, 0);
  gemm_contract<0>(lane, wv, sH2, sC[wv], W.w3a, W.b3a,
                   sT00, sT01, sT10, sT11, sOut);
  trunk(tid, lane, wv, sFeat, sH1, sH2, sMu, sRs, W, 2);
  gemm_contract<2>(lane, wv, sH2, sC[wv],
                   W.w3a + 2L * MIDC * (HIDC * DIMC), W.b3a + 2 * HIDC * DIMC,
                   sT00, sT01, sT10, sT11, sOut);
  __syncthreads();
  for (int idx = tid; idx < JJ * HIDC; idx += 256) {
    const int e = idx >> 7, o = idx & 127;
    o0g[(node * JJ + e) * HIDC + o] = sOut[e][o];
  }
  __syncthreads();

  // ---- degree-1 output: nets (0,1) and (1,1) ----------------------------
  for (int idx = tid; idx < JJ * 384; idx += 256) (&sOut[0][0])[idx] = 0.f;
  trunk(tid, lane, wv, sFeat, sH1, sH2, sMu, sRs, W, 1);
  gemm_contract<1>(lane, wv, sH2, sC[wv],
                   W.w3a + 1L * MIDC * (HIDC * DIMC), W.b3a + 1 * HIDC * DIMC,
                   sT00, sT01, sT10, sT11, sOut);
  trunk(tid, lane, wv, sFeat, sH1, sH2, sMu, sRs, W, 3);
  gemm_contract<3>(lane, wv, sH2, sC[wv], W.w3b, W.b3b,
                   sT00, sT01, sT10, sT11, sOut);
  __syncthreads();
  for (int idx = tid; idx < JJ * 384; idx += 256) {
    const int e = idx / 384, r = idx % 384;   // r = o*3 + p
    o1g[(node * JJ + e) * HIDC * 3 + r] = sOut[e][r];
  }
}

__global__ __launch_bounds__(128) void attn_kernel(
    const float* __restrict__ x0, const float* __restrict__ x1,
    const unsigned char* __restrict__ mask,
    const float* __restrict__ wq0, const float* __restrict__ wq1,
    const float* __restrict__ wo0, const float* __restrict__ wo1,
    const float* __restrict__ ws, float* __restrict__ out) {
  const int node = blockIdx.x;
  const int tid  = threadIdx.x;
  const float* k0 = ws + K0_OFF;
  const float* k1 = ws + K1_OFF;
  const float* v0 = ws + V0_OFF;
  const float* v1 = ws + V1_OFF;

  __shared__ float q0[HIDC], q1[HIDC][3];
  __shared__ float s0[4][16], s1[4][16];
  __shared__ float o0s[HIDC], o1s[HIDC][3];

  {
    const int o = tid;
    float a = 0.f;
    for (int d = 0; d < DIMC; ++d) a += x0[node * DIMC + d] * wq0[d * HIDC + o];
    q0[o] = a;
    for (int m = 0; m < 3; ++m) {
      float s = 0.f;
      for (int d = 0; d < DIMC; ++d)
        s += x1[(node * DIMC + d) * 3 + m] * wq1[d * HIDC + o];
      q1[o][m] = s;
    }
  }
  __syncthreads();
  if (tid < 64) {
    const int h = tid >> 4, j = tid & 15;
    const bool mk = mask[node * JJ + j] != 0;
    float a = 0.f, b = 0.f;
    for (int d = 0; d < 32; ++d) {
      const int o = h * 32 + d;
      a += q0[o] * k0[(node * JJ + j) * HIDC + o];
      for (int m = 0; m < 3; ++m)
        b += q1[o][m] * k1[((node * JJ + j) * HIDC + o) * 3 + m];
    }
    const float sc = 0.17677669529663687f;   // 32^-0.5
    s0[h][j] = mk ? a * sc : -3.0e38f;
    s1[h][j] = mk ? b * sc : -3.0e38f;
  }
  __syncthreads();
  if (tid < 8) {
    float* row = (tid < 4) ? s0[tid] : s1[tid - 4];
    float mx = row[0];
    for (int j = 1; j < 16; ++j) mx = fmaxf(mx, row[j]);
    float sm = 0.f;
    for (int j = 0; j < 16; ++j) { float e = __expf(row[j] - mx); row[j] = e; sm += e; }
    const float inv = 1.f / sm;
    for (int j = 0; j < 16; ++j) row[j] *= inv;
  }
  __syncthreads();
  {
    const int o = tid, h = o >> 5;
    float a = 0.f;
    for (int j = 0; j < 16; ++j) a += s0[h][j] * v0[(node * JJ + j) * HIDC + o];
    o0s[o] = a;
    for (int m = 0; m < 3; ++m) {
      float s = 0.f;
      for (int j = 0; j < 16; ++j)
        s += s1[h][j] * v1[((node * JJ + j) * HIDC + o) * 3 + m];
      o1s[o][m] = s;
    }
  }
  __syncthreads();
  if (tid < DIMC) {
    const int d = tid;
    float r = 0.f;
    for (int o = 0; o < HIDC; ++o) r += o0s[o] * wo0[o * DIMC + d];
    out[node * DIMC + d] = r;
    for (int m = 0; m < 3; ++m) {
      float r1 = 0.f;
      for (int o = 0; o < HIDC; ++o) r1 += o1s[o][m] * wo1[o * DIMC + d];
      out[NODES * DIMC + (node * DIMC + d) * 3 + m] = r1;
    }
  }
}

}  // namespace

extern "C" void kernel_launch(void* const* d_in, const int* in_sizes, int n_in,
                              void* d_out, int out_size, void* d_ws,
                              size_t ws_size, hipStream_t stream) {
  (void)in_sizes; (void)n_in; (void)out_size; (void)ws_size;
  const float* x0   = (const float*)d_in[0];
  const float* x1   = (const float*)d_in[1];
  const int*   nidx = (const int*)d_in[2];
  const unsigned char* mask = (const unsigned char*)d_in[3];  // bool array
  const float* rel  = (const float*)d_in[4];
  const float* b00  = (const float*)d_in[5];
  const float* b01  = (const float*)d_in[6];
  const float* b10  = (const float*)d_in[7];
  const float* b11  = (const float*)d_in[8];

  ConvW Wk, Wv;
  {
    const float** f = (const float**)&Wk;
    for (int i = 0; i < 12; ++i) f[i] = (const float*)d_in[9 + i];
    const float** g = (const float**)&Wv;
    for (int i = 0; i < 12; ++i) g[i] = (const float*)d_in[21 + i];
  }
  const float* wq0 = (const float*)d_in[33];
  const float* wq1 = (const float*)d_in[34];
  const float* wo0 = (const float*)d_in[35];
  const float* wo1 = (const float*)d_in[36];

  float* ws = (float*)d_ws;

  conv_kernel<<<NODES * 2, 256, 0, stream>>>(x0, x1, nidx, rel, b00, b01, b10,
                                             b11, Wk, Wv, ws);
  attn_kernel<<<NODES, 128, 0, stream>>>(x0, x1, mask, wq0, wq1, wo0, wo1, ws,
                                         (float*)d_out);
}